// TransformerInterpNetwork_21174188770103
// MI455X (gfx1250) — compile-verified
//
#include <hip/hip_runtime.h>
#include <hip/hip_bf16.h>

#define T_ 16
#define D_ 192
#define H_ 128
#define P_ 128
#define HEADS_ 4
#define HD_ 32
#define EC_ 32
#define S_ (T_*D_)          // 3072
#define TM1_ (T_-1)         // 15

typedef __attribute__((ext_vector_type(2))) float v2f;
typedef __attribute__((ext_vector_type(8))) float v8f;

#define EPI_BIAS 1
#define EPI_RELU 2
#define EPI_ACC  4

// Use CDNA5 async global->LDS DMA path (GLOBAL_LOAD_ASYNC_TO_LDS_B128,
// tracked by ASYNCcnt) for GEMM B-tile staging. Issued via inline asm since
// the clang builtins differ across toolchains.
#define USE_ASYNC_LDS 1

__device__ __forceinline__ v8f wmma_f32(v2f a, v2f b, v8f c) {
    return __builtin_amdgcn_wmma_f32_16x16x4_f32(false, a, false, b, (short)0, c, false, false);
}

__device__ __forceinline__ v8f vzero8() {
    v8f v = {0.f,0.f,0.f,0.f,0.f,0.f,0.f,0.f};
    return v;
}

#if USE_ASYNC_LDS
__device__ __forceinline__ void async_copy_b128(const float* gsrc, const float* lds_dst) {
    // VDST = LDS byte address (low 32 bits of the flat LDS-aperture address),
    // VADDR = 64-bit global address, no SADDR.
    unsigned lds_off = (unsigned)(size_t)lds_dst;
    asm volatile("global_load_async_to_lds_b128 %0, %1, off"
                 :: "v"(lds_off), "v"(gsrc) : "memory");
}
__device__ __forceinline__ void wait_asynccnt0() {
    asm volatile("s_wait_asynccnt 0" ::: "memory");
}
#endif

// ---------------------------------------------------------------------------
// Generic WMMA f32 GEMM: C[M,N] (= or +=) A[M,K](lda) @ B[K,N](ldb) [+bias][relu]
// block = 256 threads = 8 waves; wave w -> rows [bx*128+16w, +16), cols [by*16, +16)
// K <= 512, M % 16 == 0, N % 16 == 0, ldb % 4 == 0 (16B alignment for async).
// ---------------------------------------------------------------------------
__global__ __launch_bounds__(256) void gemm_wmma_generic(
    const float* __restrict__ A, int lda,
    const float* __restrict__ B, int ldb,
    const float* __restrict__ bias,
    float* __restrict__ C, int ldc,
    int M, int N, int K, int flags)
{
    __shared__ float Bt[512*16];
    int wave = threadIdx.x >> 5, lane = threadIdx.x & 31;
    int li = lane & 15, half = lane >> 4;
    int cbase = blockIdx.y * 16;

#if USE_ASYNC_LDS
    // Stage B tile (K x 16) with async global->LDS B128 copies (4 floats/op).
    for (int idx = threadIdx.x; idx < K*4; idx += 256) {
        int k = idx >> 2, part = idx & 3;
        async_copy_b128(B + (long)k*ldb + cbase + part*4, &Bt[k*16 + part*4]);
    }
    wait_asynccnt0();
    __syncthreads();
#else
    for (int idx = threadIdx.x; idx < K*16; idx += 256) {
        int k = idx >> 4, j = idx & 15;
        Bt[idx] = B[(long)k*ldb + cbase + j];
    }
    __syncthreads();
#endif

    int rbase = blockIdx.x * 128 + wave * 16;
    if (rbase >= M) return;

    const float* Arow = A + (long)(rbase + li) * lda;
    v8f acc = vzero8();
    for (int k0 = 0; k0 < K; k0 += 4) {
        int ka = k0 + 2*half;
        v2f a; a.x = Arow[ka]; a.y = Arow[ka+1];
        v2f b; b.x = Bt[ka*16 + li]; b.y = Bt[(ka+1)*16 + li];
        acc = wmma_f32(a, b, acc);
    }
    for (int r = 0; r < 8; ++r) {
        int row = rbase + r + 8*half;
        int col = cbase + li;
        float v = acc[r];
        if (flags & EPI_BIAS) v += bias[col];
        if (flags & EPI_ACC)  v += C[(long)row*ldc + col];
        if (flags & EPI_RELU) v = fmaxf(v, 0.f);
        C[(long)row*ldc + col] = v;
    }
}

// ---------------------------------------------------------------------------
// Node embedding: z[s,p] = relu( sum_h x[t,h,d]*W_node[h,p] + b_node[p] ) + time_enc[p]*t
// s = t*D + d.  M=S, K=H, N=P.
// ---------------------------------------------------------------------------
__global__ __launch_bounds__(256) void k_node_embed(
    const float* __restrict__ x, const float* __restrict__ Wn,
    const float* __restrict__ bn, const float* __restrict__ tenc,
    float* __restrict__ z)
{
    __shared__ float Bt[H_*16];
    int wave = threadIdx.x >> 5, lane = threadIdx.x & 31;
    int li = lane & 15, half = lane >> 4;
    int cbase = blockIdx.y * 16;

    for (int idx = threadIdx.x; idx < H_*16; idx += 256) {
        int k = idx >> 4, j = idx & 15;
        Bt[idx] = Wn[k*P_ + cbase + j];
    }
    __syncthreads();

    int rbase = blockIdx.x * 128 + wave * 16;
    int s = rbase + li;
    int t = s / D_, d = s % D_;
    const float* xbase = x + (long)t * H_ * D_ + d;

    v8f acc = vzero8();
    for (int k0 = 0; k0 < H_; k0 += 4) {
        int ka = k0 + 2*half;
        v2f a; a.x = xbase[(long)ka * D_]; a.y = xbase[(long)(ka+1) * D_];
        v2f b; b.x = Bt[ka*16 + li]; b.y = Bt[(ka+1)*16 + li];
        acc = wmma_f32(a, b, acc);
    }
    for (int r = 0; r < 8; ++r) {
        int row = rbase + r + 8*half;
        int col = cbase + li;
        float v = fmaxf(acc[r] + bn[col], 0.f);
        v += tenc[col] * (float)(row / D_);
        z[(long)row*P_ + col] = v;
    }
}

// ---------------------------------------------------------------------------
// Edge encoder GEMM: he[m,p] = sum_c relu(edge_w[m]*We[c]+be[c]) * Bw[c,p]
// M = D*D, K = EC = 32, N = P. A computed on the fly.
// ---------------------------------------------------------------------------
__global__ __launch_bounds__(256) void k_he(
    const float* __restrict__ edge_w, const float* __restrict__ We,
    const float* __restrict__ be, const float* __restrict__ Bw,
    float* __restrict__ he)
{
    __shared__ float Bt[EC_*16];
    __shared__ float WeS[EC_], beS[EC_];
    int wave = threadIdx.x >> 5, lane = threadIdx.x & 31;
    int li = lane & 15, half = lane >> 4;
    int cbase = blockIdx.y * 16;

    for (int idx = threadIdx.x; idx < EC_*16; idx += 256) {
        int k = idx >> 4, j = idx & 15;
        Bt[idx] = Bw[k*P_ + cbase + j];
    }
    if (threadIdx.x < EC_) { WeS[threadIdx.x] = We[threadIdx.x]; beS[threadIdx.x] = be[threadIdx.x]; }
    __syncthreads();

    int rbase = blockIdx.x * 128 + wave * 16;
    float ew = edge_w[rbase + li];

    v8f acc = vzero8();
    for (int k0 = 0; k0 < EC_; k0 += 4) {
        int ka = k0 + 2*half;
        v2f a;
        a.x = fmaxf(ew * WeS[ka]   + beS[ka],   0.f);
        a.y = fmaxf(ew * WeS[ka+1] + beS[ka+1], 0.f);
        v2f b; b.x = Bt[ka*16 + li]; b.y = Bt[(ka+1)*16 + li];
        acc = wmma_f32(a, b, acc);
    }
    for (int r = 0; r < 8; ++r) {
        int row = rbase + r + 8*half;
        he[(long)row*P_ + cbase + li] = acc[r];
    }
}

// ---------------------------------------------------------------------------
// Adjacency bitmask: adjm[d1][w] bit d2 set iff edge_w[d1,d2]>0 || d1==d2
// ---------------------------------------------------------------------------
__global__ void k_adjmask(const float* __restrict__ edge_w, unsigned* __restrict__ adjm)
{
    int d1 = threadIdx.x;
    if (d1 >= D_) return;
    unsigned w[6] = {0,0,0,0,0,0};
    for (int d2 = 0; d2 < D_; ++d2) {
        bool a = (edge_w[d1*D_ + d2] > 0.f) || (d2 == d1);
        if (a) w[d2 >> 5] |= (1u << (d2 & 31));
    }
    for (int i = 0; i < 6; ++i) adjm[d1*6 + i] = w[i];
}

// ---------------------------------------------------------------------------
// LayerNorm over last dim (P=128): one wave per row, 8 rows per block.
// ---------------------------------------------------------------------------
__global__ __launch_bounds__(256) void k_ln(
    const float* __restrict__ z, const float* __restrict__ g,
    const float* __restrict__ b, float* __restrict__ y)
{
    int row = blockIdx.x * 8 + (threadIdx.x >> 5);
    int lane = threadIdx.x & 31;
    const float* zr = z + (long)row * P_;
    float v[4], s = 0.f;
    for (int i = 0; i < 4; ++i) { v[i] = zr[lane + 32*i]; s += v[i]; }
    for (int off = 1; off < 32; off <<= 1) s += __shfl_xor(s, off, 32);
    float mean = s * (1.0f / P_);
    float q = 0.f;
    for (int i = 0; i < 4; ++i) { float d = v[i] - mean; q += d*d; }
    for (int off = 1; off < 32; off <<= 1) q += __shfl_xor(q, off, 32);
    float rstd = rsqrtf(q * (1.0f / P_) + 1e-5f);
    float* yr = y + (long)row * P_;
    for (int i = 0; i < 4; ++i) {
        int col = lane + 32*i;
        yr[col] = (v[i] - mean) * rstd * g[col] + b[col];
    }
}

// ---------------------------------------------------------------------------
// Block-sparse flash attention. Block = (t1, head). Keys = times {t1, t1-1}
// processed as chunks of 192 staged in LDS. Online softmax, WMMA scores & P@V.
// ---------------------------------------------------------------------------
__global__ __launch_bounds__(256) void k_attn(
    const float* __restrict__ qkv, const unsigned* __restrict__ adjm,
    float* __restrict__ attnout)
{
    int t1 = blockIdx.x >> 2, hh = blockIdx.x & 3;
    int wave = threadIdx.x >> 5, lane = threadIdx.x & 31;
    int li = lane & 15, half = lane >> 4;

    __shared__ float kT[HD_][D_ + 1];       // 32 x 193 (B operand for scores)
    __shared__ float vS[D_][HD_ + 1];       // 192 x 33
    __shared__ unsigned adjS[D_*6];
    __shared__ float pbuf[8][16][17];       // per-wave C->A transpose buffer

    for (int idx = threadIdx.x; idx < D_*6; idx += 256) adjS[idx] = adjm[idx];

    float m[2][8], l[2][8];
    v8f o0[2], o1[2];
    for (int qi = 0; qi < 2; ++qi) {
        for (int r = 0; r < 8; ++r) { m[qi][r] = -1e30f; l[qi][r] = 0.f; }
        o0[qi] = vzero8(); o1[qi] = vzero8();
    }

    const float scale = 0.17677669529663687f;  // 1/sqrt(32)
    int nchunk = (t1 == 0) ? 1 : 2;

    for (int c = 0; c < nchunk; ++c) {
        int tk = t1 - c;
        __syncthreads();
        for (int idx = threadIdx.x; idx < D_*HD_; idx += 256) {
            int j = idx >> 5, cc = idx & 31;
            long row = (long)(tk*D_ + j) * (3*P_) + hh*HD_;
            kT[cc][j] = qkv[row + P_ + cc];
            vS[j][cc] = qkv[row + 2*P_ + cc];
        }
        __syncthreads();

        for (int qi = 0; qi < 2; ++qi) {
            int qt = wave + 8*qi;
            if (qt >= D_/16) break;
            int qbase = qt * 16;
            long qrow = (long)(t1*D_ + qbase + li) * (3*P_) + hh*HD_;
            v2f af[8];
            for (int ks = 0; ks < 8; ++ks) {
                int ka = 4*ks + 2*half;
                af[ks].x = qkv[qrow + ka];
                af[ks].y = qkv[qrow + ka + 1];
            }
            for (int jt = 0; jt < D_/16; ++jt) {
                int jbase = jt * 16;
                v8f sc = vzero8();
                for (int ks = 0; ks < 8; ++ks) {
                    int kb = 4*ks + 2*half;
                    v2f b; b.x = kT[kb][jbase+li]; b.y = kT[kb+1][jbase+li];
                    sc = wmma_f32(af[ks], b, sc);
                }
                int dk = jbase + li;
                float pv[8], rmax[8];
                for (int r = 0; r < 8; ++r) {
                    int dq = qbase + r + 8*half;
                    bool ok = (adjS[dq*6 + (dk >> 5)] >> (dk & 31)) & 1u;
                    float s = ok ? sc[r] * scale : -1e30f;
                    float mx = s;
                    for (int off = 1; off < 16; off <<= 1) mx = fmaxf(mx, __shfl_xor(mx, off, 32));
                    pv[r] = s; rmax[r] = mx;
                }
                for (int r = 0; r < 8; ++r) {
                    float mn = fmaxf(m[qi][r], rmax[r]);
                    float alpha = __expf(m[qi][r] - mn);
                    float p = __expf(pv[r] - mn);
                    float ps = p;
                    for (int off = 1; off < 16; off <<= 1) ps += __shfl_xor(ps, off, 32);
                    l[qi][r] = l[qi][r] * alpha + ps;
                    m[qi][r] = mn;
                    o0[qi][r] *= alpha; o1[qi][r] *= alpha;
                    pbuf[wave][r + 8*half][li] = p;
                }
                for (int ks = 0; ks < 4; ++ks) {
                    int kp = 4*ks + 2*half;
                    v2f a; a.x = pbuf[wave][li][kp]; a.y = pbuf[wave][li][kp+1];
                    v2f b0; b0.x = vS[jbase+kp][li];      b0.y = vS[jbase+kp+1][li];
                    v2f b1; b1.x = vS[jbase+kp][li+16];   b1.y = vS[jbase+kp+1][li+16];
                    o0[qi] = wmma_f32(a, b0, o0[qi]);
                    o1[qi] = wmma_f32(a, b1, o1[qi]);
                }
            }
        }
    }

    for (int qi = 0; qi < 2; ++qi) {
        int qt = wave + 8*qi;
        if (qt >= D_/16) break;
        int qbase = qt * 16;
        for (int r = 0; r < 8; ++r) {
            float inv = 1.0f / l[qi][r];
            int row = t1*D_ + qbase + r + 8*half;
            long orow = (long)row * P_ + hh*HD_;
            attnout[orow + li]      = o0[qi][r] * inv;
            attnout[orow + 16 + li] = o1[qi][r] * inv;
        }
    }
}

// ---------------------------------------------------------------------------
// logits[t,i,j] = relu(hn[t,i,:]+hc[t,j,:]+he[i,j,:]+bc1) . Wc2 + bc2, masked.
// Block per (t,i); 8 waves loop over j.
// ---------------------------------------------------------------------------
__global__ __launch_bounds__(256) void k_logits(
    const float* __restrict__ hn, const float* __restrict__ hc,
    const float* __restrict__ he, const float* __restrict__ bc1,
    const float* __restrict__ Wc2, const float* __restrict__ bc2,
    const float* __restrict__ edge_w, float* __restrict__ logits)
{
    int t = blockIdx.x / D_, i = blockIdx.x % D_;
    int wave = threadIdx.x >> 5, lane = threadIdx.x & 31;
    __shared__ float hnS[P_], wS[P_], bS[P_];
    for (int c = threadIdx.x; c < P_; c += 256) {
        hnS[c] = hn[((long)t*D_ + i)*P_ + c];
        wS[c] = Wc2[c];
        bS[c] = bc1[c];
    }
    __syncthreads();
    for (int j = wave; j < D_; j += 8) {
        const float* hcr = hc + ((long)t*D_ + j)*P_;
        const float* her = he + ((long)i*D_ + j)*P_;
        float s = 0.f;
        for (int c = lane; c < P_; c += 32) {
            float v = hnS[c] + hcr[c] + her[c] + bS[c];
            s += fmaxf(v, 0.f) * wS[c];
        }
        for (int off = 1; off < 32; off <<= 1) s += __shfl_xor(s, off, 32);
        if (lane == 0) {
            float ew = edge_w[i*D_ + j];
            bool masked = (ew == 0.f) && (i != j);
            logits[((long)t*D_ + i)*D_ + j] = masked ? -1e9f : (s + bc2[0]);
        }
    }
}

// ---------------------------------------------------------------------------
// dist[row] = hd[row,:] . Wd2 + bd2 ; one wave per row.
// ---------------------------------------------------------------------------
__global__ __launch_bounds__(256) void k_dist(
    const float* __restrict__ hd, const float* __restrict__ Wd2,
    const float* __restrict__ bd2, float* __restrict__ dist)
{
    int row = blockIdx.x * 8 + (threadIdx.x >> 5);
    if (row >= TM1_*D_) return;
    int lane = threadIdx.x & 31;
    float s = 0.f;
    for (int c = lane; c < P_; c += 32) s += hd[(long)row*P_ + c] * Wd2[c];
    for (int off = 1; off < 32; off <<= 1) s += __shfl_xor(s, off, 32);
    if (lane == 0) dist[row] = s + bd2[0];
}

// ---------------------------------------------------------------------------
extern "C" void kernel_launch(void* const* d_in, const int* in_sizes, int n_in,
                              void* d_out, int out_size, void* d_ws, size_t ws_size,
                              hipStream_t stream)
{
    const float* x        = (const float*)d_in[0];
    const float* edge_w   = (const float*)d_in[1];
    const float* W_node   = (const float*)d_in[2];
    const float* b_node   = (const float*)d_in[3];
    const float* W_edge   = (const float*)d_in[4];
    const float* b_edge   = (const float*)d_in[5];
    const float* time_enc = (const float*)d_in[6];
    const float* Wqkv     = (const float*)d_in[7];
    const float* bqkv     = (const float*)d_in[8];
    const float* Wo       = (const float*)d_in[9];
    const float* bo       = (const float*)d_in[10];
    const float* W1       = (const float*)d_in[11];
    const float* b1       = (const float*)d_in[12];
    const float* W2       = (const float*)d_in[13];
    const float* b2       = (const float*)d_in[14];
    const float* g1       = (const float*)d_in[15];
    const float* beta1    = (const float*)d_in[16];
    const float* g2       = (const float*)d_in[17];
    const float* beta2    = (const float*)d_in[18];
    const float* Wc1      = (const float*)d_in[19];
    const float* bc1      = (const float*)d_in[20];
    const float* Wc2      = (const float*)d_in[21];
    const float* bc2      = (const float*)d_in[22];
    const float* Wd1      = (const float*)d_in[23];
    const float* bd1      = (const float*)d_in[24];
    const float* Wd2      = (const float*)d_in[25];
    const float* bd2      = (const float*)d_in[26];

    float* ws      = (float*)d_ws;
    float* z       = ws;                                   // S*P
    float* y       = z + (size_t)S_*P_;                    // S*P
    float* qkv     = y + (size_t)S_*P_;                    // S*3P
    float* attnout = qkv + (size_t)S_*3*P_;                // S*P
    float* ffn     = attnout + (size_t)S_*P_;              // S*4P
    float* he      = ffn + (size_t)S_*4*P_;                // D*D*P
    float* hn      = he + (size_t)D_*D_*P_;                // 15*D*P
    float* hc      = hn + (size_t)TM1_*D_*P_;              // 15*D*P
    float* hd      = hc + (size_t)TM1_*D_*P_;              // 15*D*P
    unsigned* adjm = (unsigned*)(hd + (size_t)TM1_*D_*P_); // 192*6 words

    float* logits = (float*)d_out;                         // 15*D*D
    float* dist   = logits + (size_t)TM1_*D_*D_;           // 15*D

    k_adjmask<<<1, 256, 0, stream>>>(edge_w, adjm);
    k_node_embed<<<dim3(S_/128, P_/16), 256, 0, stream>>>(x, W_node, b_node, time_enc, z);

    for (int it = 0; it < 2; ++it) {
        k_ln<<<S_/8, 256, 0, stream>>>(z, g1, beta1, y);
        gemm_wmma_generic<<<dim3(S_/128, 3*P_/16), 256, 0, stream>>>(
            y, P_, Wqkv, 3*P_, bqkv, qkv, 3*P_, S_, 3*P_, P_, EPI_BIAS);
        k_attn<<<T_*HEADS_, 256, 0, stream>>>(qkv, adjm, attnout);
        gemm_wmma_generic<<<dim3(S_/128, P_/16), 256, 0, stream>>>(
            attnout, P_, Wo, P_, bo, z, P_, S_, P_, P_, EPI_BIAS|EPI_ACC);
        k_ln<<<S_/8, 256, 0, stream>>>(z, g2, beta2, y);
        gemm_wmma_generic<<<dim3(S_/128, 4*P_/16), 256, 0, stream>>>(
            y, P_, W1, 4*P_, b1, ffn, 4*P_, S_, 4*P_, P_, EPI_BIAS|EPI_RELU);
        gemm_wmma_generic<<<dim3(S_/128, P_/16), 256, 0, stream>>>(
            ffn, 4*P_, W2, P_, b2, z, P_, S_, P_, 4*P_, EPI_BIAS|EPI_ACC);
    }

    int Mtail = TM1_*D_;                 // 2880
    int gx = (Mtail + 127) / 128;        // 23
    // hn = nxt @ Wc1[:P] ; hc = curr @ Wc1[P:2P]
    gemm_wmma_generic<<<dim3(gx, P_/16), 256, 0, stream>>>(
        z + D_*P_, P_, Wc1, P_, nullptr, hn, P_, Mtail, P_, P_, 0);
    gemm_wmma_generic<<<dim3(gx, P_/16), 256, 0, stream>>>(
        z, P_, Wc1 + P_*P_, P_, nullptr, hc, P_, Mtail, P_, P_, 0);
    // he = relu(edge_w x W_edge + b_edge) @ Wc1[2P:]
    k_he<<<dim3(D_*D_/128, P_/16), 256, 0, stream>>>(edge_w, W_edge, b_edge, Wc1 + 2*P_*P_, he);
    // hd = relu(curr @ Wd1[:P] + nxt @ Wd1[P:] + bd1)  (two passes)
    gemm_wmma_generic<<<dim3(gx, P_/16), 256, 0, stream>>>(
        z, P_, Wd1, P_, nullptr, hd, P_, Mtail, P_, P_, 0);
    gemm_wmma_generic<<<dim3(gx, P_/16), 256, 0, stream>>>(
        z + D_*P_, P_, Wd1 + P_*P_, P_, bd1, hd, P_, Mtail, P_, P_, EPI_BIAS|EPI_ACC|EPI_RELU);

    k_logits<<<TM1_*D_, 256, 0, stream>>>(hn, hc, he, bc1, Wc2, bc2, edge_w, logits);
    k_dist<<<(Mtail + 7)/8, 256, 0, stream>>>(hd, Wd2, bd2, dist);
}